// JointAttention_42992622633024
// MI455X (gfx1250) — compile-verified
//
#include <hip/hip_runtime.h>
#include <math.h>

// ---------------------------------------------------------------------------
// JointAttention on MI455X (gfx1250, wave32, WMMA + Tensor Data Mover).
// In the reference, out[:, :HEADS] keeps only the x-derived heads, so the
// entire y pathway (y, gy, by, Wqkv_y) is dead code. We compute:
//   xn = LN(x); qkv = xn @ Wqkv_x; xo = attention over 16 x-heads;
//   out_x = xo @ Wout_x + bout_x; out_y = xo @ Wout_y + bout_y.
// Precision: f32 WMMA (V_WMMA_F32_16X16X4_F32) — ~60 GFLOP / ~70MB fits in
// L2; no roofline payoff for dropping precision, keep exact f32 semantics.
// Data movement: TDM (tensor_load_to_lds) double-buffers LDS tiles so the
// DMA of tile k+1 overlaps the WMMA loop on tile k (s_wait_tensorcnt gate).
// ---------------------------------------------------------------------------

typedef __attribute__((ext_vector_type(2))) float v2f;
typedef __attribute__((ext_vector_type(8))) float v8f;
typedef __attribute__((ext_vector_type(4))) unsigned int v4u;
typedef __attribute__((ext_vector_type(8))) int v8i;
typedef __attribute__((ext_vector_type(4))) int v4i;

#define DIMX     1024
#define HEADS    16
#define DIM_HEAD 64
#define INNER    1024
#define BATCH    4
#define SEQ      1024
#define TOKENS   (BATCH * SEQ)       // 4096
#define QK_SCALE 0.125f              // 64^-0.5

static __device__ __forceinline__ v8f v8f_zero() {
  v8f z;
#pragma unroll
  for (int i = 0; i < 8; ++i) z[i] = 0.0f;
  return z;
}

static __device__ __forceinline__ v8f wmma_f32(v2f a, v2f b, v8f c) {
  // D = A(16x4) * B(4x16) + C(16x16), all f32, wave32.
  return __builtin_amdgcn_wmma_f32_16x16x4_f32(
      /*neg_a=*/false, a, /*neg_b=*/false, b,
      /*c_mod=*/(short)0, c, /*reuse_a=*/false, /*reuse_b=*/false);
}

// Low 32 bits of a generic pointer to LDS == LDS byte address (ISA aperture
// rule).
static __device__ __forceinline__ unsigned lds_off(const void* p) {
  return (unsigned)(unsigned long long)p;
}

// ---------------------------------------------------------------------------
// TDM: async DMA of a 2D f32 tile (tile_w x tile_h, row stride in elements)
// from global memory into LDS. Descriptor per CDNA5 ISA 10.8 (D# groups).
// tensor_dim = tile_dim (tiles never go OOB here). Tracked by TENSORcnt.
// 6-arg builtin form (clang-23 / therock-10.0 headers).
// ---------------------------------------------------------------------------
static __device__ __forceinline__ void tdm_load_2d(unsigned lds_byte_addr,
                                                   const void* gptr,
                                                   unsigned tile_w,
                                                   unsigned tile_h,
                                                   unsigned row_stride_elems) {
  unsigned long long ga = (unsigned long long)gptr;
  v4u g0;
  g0[0] = 1u;                                        // count=1, user mode
  g0[1] = lds_byte_addr;                             // lds_addr [63:32]
  g0[2] = (unsigned)(ga & 0xFFFFFFFFu);              // global_addr lo
  g0[3] = (unsigned)((ga >> 32) & 0x01FFFFFFu)       // global_addr [56:32]
          | (2u << 30);                              // type=2 ("image")
  v8i g1;
  g1[0] = (int)(2u << 16);                           // data_size=2 -> 4B; mask=0
  g1[1] = (int)((tile_w & 0xFFFFu) << 16);           // tensor_dim0[15:0]
  g1[2] = (int)(((tile_w >> 16) & 0xFFFFu)           // tensor_dim0[31:16]
          | ((tile_h & 0xFFFFu) << 16));             // tensor_dim1[15:0]
  g1[3] = (int)(((tile_h >> 16) & 0xFFFFu)           // tensor_dim1[31:16]
          | ((tile_w & 0xFFFFu) << 16));             // tile_dim0
  g1[4] = (int)(tile_h & 0xFFFFu);                   // tile_dim1; tile_dim2=0
  g1[5] = (int)row_stride_elems;                     // tensor_dim0_stride lo
  g1[6] = 0;                                         // stride hi, dim1_stride
  g1[7] = 0;
  v4i gz4;
  gz4[0] = 0; gz4[1] = 0; gz4[2] = 0; gz4[3] = 0;    // groups 2/3 unused (2D)
  v8i gz8;
#pragma unroll
  for (int i = 0; i < 8; ++i) gz8[i] = 0;
  __builtin_amdgcn_tensor_load_to_lds(g0, g1, gz4, gz4, gz8, 0);
}

// ---------------------------------------------------------------------------
// LayerNorm: one block (256 threads) per row of 1024.
// ---------------------------------------------------------------------------
__global__ __launch_bounds__(256) void ln_kernel(const float* __restrict__ x,
                                                 const float* __restrict__ g,
                                                 const float* __restrict__ b,
                                                 float* __restrict__ out) {
  const int row = blockIdx.x;
  const float* xr = x + (size_t)row * DIMX;
  float* orow = out + (size_t)row * DIMX;

  float vals[4];
  float s = 0.f, ss = 0.f;
#pragma unroll
  for (int i = 0; i < 4; ++i) {
    float v = xr[threadIdx.x + i * 256];
    vals[i] = v;
    s += v;
    ss += v * v;
  }
  __shared__ float red0[256];
  __shared__ float red1[256];
  red0[threadIdx.x] = s;
  red1[threadIdx.x] = ss;
  __syncthreads();
  for (int off = 128; off > 0; off >>= 1) {
    if (threadIdx.x < off) {
      red0[threadIdx.x] += red0[threadIdx.x + off];
      red1[threadIdx.x] += red1[threadIdx.x + off];
    }
    __syncthreads();
  }
  const float mu = red0[0] * (1.0f / DIMX);
  const float var = red1[0] * (1.0f / DIMX) - mu * mu;
  const float inv = rsqrtf(var + 1e-5f);
#pragma unroll
  for (int i = 0; i < 4; ++i) {
    int c = threadIdx.x + i * 256;
    orow[c] = (vals[i] - mu) * inv * g[c] + b[c];
  }
}

// ---------------------------------------------------------------------------
// GEMM: C[M,N] = A[M,K] @ W[K,N] (+ bias). M,N multiples of 64, K of 32.
// Block: 128 threads = 4 wave32; tile 64x64, BK=32, double-buffered LDS.
// Wave 0 issues TDM loads for tile k+1 while all waves run WMMA on tile k;
// s_wait_tensorcnt 0 + workgroup barrier hand the new tile over.
// Fragment mapping per CDNA5 ISA 7.12.2 (32-bit A 16x4, 32-bit C/D 16x16).
// ---------------------------------------------------------------------------
__global__ __launch_bounds__(128) void gemm_kernel(const float* __restrict__ A,
                                                   const float* __restrict__ W,
                                                   const float* __restrict__ bias,
                                                   float* __restrict__ C,
                                                   int M, int N, int K) {
  __shared__ float As[2][64][32];   // 16 KB
  __shared__ float Ws[2][32][64];   // 16 KB

  const int cn0 = blockIdx.x * 64;
  const int rm0 = blockIdx.y * 64;
  const int tid = threadIdx.x;
  const int lane = tid & 31;
  const int wave = tid >> 5;
  const int m = lane & 15;           // A row / B col / D col within tile
  const int hi = (lane >> 4) & 1;    // half-wave select
  const int koff = hi * 2;           // A/B K-offset for upper half-wave
  const bool loader = (wave == 0);

  v8f acc[4];
#pragma unroll
  for (int nt = 0; nt < 4; ++nt) acc[nt] = v8f_zero();

  // Prologue: DMA first K-tile into buffer 0.
  if (loader) {
    tdm_load_2d(lds_off(&As[0][0][0]), A + (size_t)rm0 * K, 32, 64, K);
    tdm_load_2d(lds_off(&Ws[0][0][0]), W + cn0, 64, 32, N);
  }
  __builtin_amdgcn_s_wait_tensorcnt(0);
  __syncthreads();

  int cur = 0;
  for (int k0 = 0; k0 < K; k0 += 32) {
    // Kick off DMA of the next K-tile into the other buffer (overlaps WMMA).
    if (loader && (k0 + 32 < K)) {
      tdm_load_2d(lds_off(&As[cur ^ 1][0][0]),
                  A + (size_t)rm0 * K + (k0 + 32), 32, 64, K);
      tdm_load_2d(lds_off(&Ws[cur ^ 1][0][0]),
                  W + (size_t)(k0 + 32) * N + cn0, 64, 32, N);
    }

    const int rw = wave * 16;
#pragma unroll
    for (int kk = 0; kk < 32; kk += 4) {
      // A frag: lanes 0-15 -> K = kk,kk+1 ; lanes 16-31 -> K = kk+2,kk+3.
      float2 af = *(const float2*)&As[cur][rw + m][kk + koff];
      v2f a; a.x = af.x; a.y = af.y;
#pragma unroll
      for (int nt = 0; nt < 4; ++nt) {
        v2f bfr;
        bfr.x = Ws[cur][kk + koff][nt * 16 + m];
        bfr.y = Ws[cur][kk + koff + 1][nt * 16 + m];
        acc[nt] = wmma_f32(a, bfr, acc[nt]);
      }
    }

    __builtin_amdgcn_s_wait_tensorcnt(0);  // wave0: next tile landed in LDS
    __syncthreads();
    cur ^= 1;
  }

  // D layout: register r -> row (r + hi*8), lane -> col m within 16-col tile.
#pragma unroll
  for (int nt = 0; nt < 4; ++nt) {
    int col = cn0 + nt * 16 + m;
    float bv = bias ? bias[col] : 0.0f;
#pragma unroll
    for (int r = 0; r < 8; ++r) {
      int row = rm0 + wave * 16 + r + hi * 8;
      C[(size_t)row * N + col] = acc[nt][r] + bv;
    }
  }
}

// ---------------------------------------------------------------------------
// Flash attention, one block per (batch, head, 64-query chunk).
// 128 threads = 4 waves; wave w owns 16 query rows. K/V streamed in 64-key
// tiles via TDM with double buffering; running-max/denominator softmax;
// P staged per-wave in LDS to convert D-layout -> A-layout for the PV WMMA.
// qkv layout: [token(4096)][3*INNER]; q at +0, k at +INNER, v at +2*INNER.
// xo layout:  [token(4096)][INNER].
// ---------------------------------------------------------------------------
__global__ __launch_bounds__(128) void attn_kernel(const float* __restrict__ qkv,
                                                   float* __restrict__ xo) {
  __shared__ float Ks[2][64][64];    // 32 KB, [key][d]
  __shared__ float Vs[2][64][64];    // 32 KB, [key][d]
  __shared__ float Ps[4][16][64];    // 16 KB, per-wave P tile [qrow][key]

  const int idx = blockIdx.x;
  const int b = idx >> 8;            // / (16 heads * 16 qblocks)
  const int h = (idx >> 4) & 15;
  const int qb = idx & 15;

  const int tid = threadIdx.x;
  const int lane = tid & 31;
  const int wave = tid >> 5;
  const int m = lane & 15;
  const int hi = (lane >> 4) & 1;
  const int koff = hi * 2;
  const bool loader = (wave == 0);

  // Q fragments for this wave's 16 rows live in registers: A-layout is a
  // natural strided global read (row = lane%16, K-pair selected by half-wave).
  v2f qf[16];
  {
    const int qrow = b * SEQ + qb * 64 + wave * 16 + m;
    const float* qptr = qkv + (size_t)qrow * (3 * INNER) + h * DIM_HEAD;
#pragma unroll
    for (int kk = 0; kk < 16; ++kk) {
      float2 t = *(const float2*)(qptr + kk * 4 + koff);
      qf[kk].x = t.x;
      qf[kk].y = t.y;
    }
  }

  v8f o[4];
#pragma unroll
  for (int nt = 0; nt < 4; ++nt) o[nt] = v8f_zero();
  float m_run[8], l_run[8];
#pragma unroll
  for (int r = 0; r < 8; ++r) { m_run[r] = -1e30f; l_run[r] = 0.0f; }

  const float* kvbase = qkv + (size_t)(b * SEQ) * (3 * INNER) + h * DIM_HEAD;

  // Prologue: DMA first K/V tiles into buffer 0.
  if (loader) {
    tdm_load_2d(lds_off(&Ks[0][0][0]), kvbase + INNER, 64, 64, 3 * INNER);
    tdm_load_2d(lds_off(&Vs[0][0][0]), kvbase + 2 * INNER, 64, 64, 3 * INNER);
  }
  __builtin_amdgcn_s_wait_tensorcnt(0);
  __syncthreads();

  int cur = 0;
  for (int kb = 0; kb < SEQ / 64; ++kb) {
    // DMA next key block's K/V into the other buffer while we compute.
    if (loader && (kb + 1 < SEQ / 64)) {
      const float* nb = kvbase + (size_t)(kb + 1) * 64 * (3 * INNER);
      tdm_load_2d(lds_off(&Ks[cur ^ 1][0][0]), nb + INNER, 64, 64, 3 * INNER);
      tdm_load_2d(lds_off(&Vs[cur ^ 1][0][0]), nb + 2 * INNER, 64, 64, 3 * INNER);
    }

    // S = Q @ K^T : B[d][key] = Ks[key][d] (transposed LDS read).
    v8f s[4];
#pragma unroll
    for (int nt = 0; nt < 4; ++nt) s[nt] = v8f_zero();
#pragma unroll
    for (int kk = 0; kk < 16; ++kk) {
      v2f a = qf[kk];
#pragma unroll
      for (int nt = 0; nt < 4; ++nt) {
        float2 t = *(const float2*)&Ks[cur][nt * 16 + m][kk * 4 + koff];
        v2f bfr; bfr.x = t.x; bfr.y = t.y;
        s[nt] = wmma_f32(a, bfr, s[nt]);
      }
    }

    // Online softmax per D-register row r (row = r + hi*8; lanes of each
    // 16-lane half hold the 64 key columns across the 4 n-tiles).
    float alpha[8];
#pragma unroll
    for (int r = 0; r < 8; ++r) {
      float mx = -1e30f;
#pragma unroll
      for (int nt = 0; nt < 4; ++nt) mx = fmaxf(mx, s[nt][r] * QK_SCALE);
      // Reduce within each 16-lane half (masks 1..8 stay inside the half).
#pragma unroll
      for (int msk = 1; msk < 16; msk <<= 1) mx = fmaxf(mx, __shfl_xor(mx, msk, 32));
      float mnew = fmaxf(m_run[r], mx);
      alpha[r] = __expf(m_run[r] - mnew);
      float rsum = 0.0f;
#pragma unroll
      for (int nt = 0; nt < 4; ++nt) {
        float p = __expf(s[nt][r] * QK_SCALE - mnew);
        s[nt][r] = p;
        rsum += p;
      }
#pragma unroll
      for (int msk = 1; msk < 16; msk <<= 1) rsum += __shfl_xor(rsum, msk, 32);
      l_run[r] = l_run[r] * alpha[r] + rsum;
      m_run[r] = mnew;
    }

    // Rescale O and stage P (D-layout -> LDS row-major per wave).
#pragma unroll
    for (int nt = 0; nt < 4; ++nt) {
#pragma unroll
      for (int r = 0; r < 8; ++r) {
        o[nt][r] *= alpha[r];
        Ps[wave][r + hi * 8][nt * 16 + m] = s[nt][r];
      }
    }
    // Same-wave DS ops are in-order; no cross-wave readers of Ps[wave].

    // O += P @ V  (A = P 16x64, B = V[key][d]).
#pragma unroll
    for (int kk = 0; kk < 16; ++kk) {
      float2 t = *(const float2*)&Ps[wave][m][kk * 4 + koff];
      v2f a; a.x = t.x; a.y = t.y;
#pragma unroll
      for (int nt = 0; nt < 4; ++nt) {
        v2f bfr;
        bfr.x = Vs[cur][kk * 4 + koff][nt * 16 + m];
        bfr.y = Vs[cur][kk * 4 + koff + 1][nt * 16 + m];
        o[nt] = wmma_f32(a, bfr, o[nt]);
      }
    }

    __builtin_amdgcn_s_wait_tensorcnt(0);  // next K/V tile fully in LDS
    __syncthreads();
    cur ^= 1;
  }

  // Normalize and write xo[token][h*64 + d].
  const int qrow0 = b * SEQ + qb * 64 + wave * 16;
#pragma unroll
  for (int nt = 0; nt < 4; ++nt) {
    int col = h * DIM_HEAD + nt * 16 + m;
#pragma unroll
    for (int r = 0; r < 8; ++r) {
      int row = qrow0 + r + hi * 8;
      xo[(size_t)row * INNER + col] = o[nt][r] / l_run[r];
    }
  }
}

// ---------------------------------------------------------------------------
// Launch: LN -> QKV GEMM -> attention -> two output GEMMs.
// Workspace: xn (16MB) | qkv (48MB) | xo (16MB) = 80MB of d_ws.
// ---------------------------------------------------------------------------
extern "C" void kernel_launch(void* const* d_in, const int* in_sizes, int n_in,
                              void* d_out, int out_size, void* d_ws, size_t ws_size,
                              hipStream_t stream) {
  const float* x      = (const float*)d_in[0];
  const float* gx     = (const float*)d_in[2];
  const float* bx     = (const float*)d_in[3];
  const float* Wqkv_x = (const float*)d_in[6];
  const float* Wout_x = (const float*)d_in[8];
  const float* bout_x = (const float*)d_in[9];
  const float* Wout_y = (const float*)d_in[10];
  const float* bout_y = (const float*)d_in[11];
  float* out = (float*)d_out;

  float* xn  = (float*)d_ws;
  float* qkv = xn + (size_t)TOKENS * DIMX;
  float* xo  = qkv + (size_t)TOKENS * 3 * INNER;

  ln_kernel<<<TOKENS, 256, 0, stream>>>(x, gx, bx, xn);

  gemm_kernel<<<dim3((3 * INNER) / 64, TOKENS / 64), 128, 0, stream>>>(
      xn, Wqkv_x, nullptr, qkv, TOKENS, 3 * INNER, DIMX);

  attn_kernel<<<BATCH * HEADS * (SEQ / 64), 128, 0, stream>>>(qkv, xo);

  gemm_kernel<<<dim3(DIMX / 64, TOKENS / 64), 128, 0, stream>>>(
      xo, Wout_x, bout_x, out, TOKENS, DIMX, INNER);
  gemm_kernel<<<dim3(DIMX / 64, TOKENS / 64), 128, 0, stream>>>(
      xo, Wout_y, bout_y, out + (size_t)TOKENS * DIMX, TOKENS, DIMX, INNER);
}